// SparseMoE_88751204205082
// MI455X (gfx1250) — compile-verified
//
#include <hip/hip_runtime.h>
#include <cstdint>

// Sparse top-2 MoE for gfx1250 using fp32 WMMA (V_WMMA_F32_16X16X4_F32).
// Pipeline: router (top-2 + gather lists) -> grouped expert GEMM (gate/up ->
// SiLU -> down, fused per 16-token tile) -> weighted combine.
// X tile staged once per block into LDS via GLOBAL_LOAD_ASYNC_TO_LDS_B128
// (ASYNCcnt-tracked), weights streamed from global (L2-resident per expert).
// Workspace layout (fp32/int32): counts[16] | lists[E*N] | wts[2N] | y[2N*D].

typedef __attribute__((ext_vector_type(2))) float v2f;
typedef __attribute__((ext_vector_type(8))) float v8f;

#define MT   16            // tokens per expert tile
#define DMAX 1024          // max supported hidden dim for LDS-resident X tile
#define SXD  (DMAX + 4)    // padded LDS stride (words) -> conflict-free A reads
#define HC   128           // H chunk per block (8 waves * 16 cols)
#define SHS  (HC + 4)

__device__ __forceinline__ v8f wmma4(v2f a, v2f b, v8f c) {
  // D = A(16x4,f32) * B(4x16,f32) + C(16x16,f32)
  return __builtin_amdgcn_wmma_f32_16x16x4_f32(
      false, a, false, b, (short)0, c, false, false);
}

// Low 32 bits of a generic pointer into __shared__ == LDS byte offset.
__device__ __forceinline__ uint32_t lds_off(const void* p) {
  return (uint32_t)(uintptr_t)p;
}

// Async DMA of 16 bytes global -> LDS, tracked by ASYNCcnt.
__device__ __forceinline__ void async_copy_b128(uint32_t lds_byte_addr,
                                                const void* gptr) {
  asm volatile("global_load_async_to_lds_b128 %0, %1, off"
               :: "v"(lds_byte_addr), "v"((uint64_t)(uintptr_t)gptr)
               : "memory");
}

__device__ __forceinline__ void wait_async0() {
  asm volatile("s_wait_asynccnt 0x0" ::: "memory");
}

__global__ void moe_zero_counts(int* counts, int E) {
  if ((int)threadIdx.x < E) counts[threadIdx.x] = 0;
}

// One wave per token: logits -> softmax -> top-2 -> normalized weights,
// append (token<<1)|slot to the chosen experts' lists.
__global__ void moe_router(const float* __restrict__ x,
                           const float* __restrict__ rw,
                           const float* __restrict__ rb,
                           int* __restrict__ counts,
                           int* __restrict__ lists,
                           float* __restrict__ wts,
                           int N, int D, int E) {
  const int lane = threadIdx.x & 31;
  const int wave = threadIdx.x >> 5;
  const int n = blockIdx.x * (blockDim.x >> 5) + wave;
  if (n >= N) return;
  const float* xr = x + (size_t)n * D;

  float logit[16];
  for (int e = 0; e < E; ++e) {
    const float* wr = rw + (size_t)e * D;
    float s = 0.f;
    for (int d = lane; d < D; d += 32) s += xr[d] * wr[d];
    for (int off = 16; off > 0; off >>= 1) s += __shfl_down(s, off);
    logit[e] = s;  // meaningful on lane 0
  }
  if (lane == 0) {
    float mx = -1e30f;
    for (int e = 0; e < E; ++e) { logit[e] += rb[e]; mx = fmaxf(mx, logit[e]); }
    float p[16], sum = 0.f;
    for (int e = 0; e < E; ++e) { p[e] = __expf(logit[e] - mx); sum += p[e]; }
    const float inv = 1.f / sum;
    int i0 = 0; float p0 = -1.f;
    for (int e = 0; e < E; ++e) { p[e] *= inv; if (p[e] > p0) { p0 = p[e]; i0 = e; } }
    int i1 = 0; float p1 = -1.f;
    for (int e = 0; e < E; ++e) if (e != i0 && p[e] > p1) { p1 = p[e]; i1 = e; }
    const float wsum = p0 + p1 + 1e-9f;
    wts[2 * n]     = p0 / wsum;
    wts[2 * n + 1] = p1 / wsum;
    int pos0 = atomicAdd(&counts[i0], 1);
    lists[(size_t)i0 * N + pos0] = (n << 1);
    int pos1 = atomicAdd(&counts[i1], 1);
    lists[(size_t)i1 * N + pos1] = (n << 1) | 1;
  }
}

// Block = 16 gathered tokens of one expert, 256 threads = 8 waves.
// Stage 0: async-DMA the full 16xD X tile into LDS (once per block).
// Stage 1: G,U = X * Wg^T, X * Wu^T (each wave owns 16 H-cols per H-chunk).
// Stage 2: acc += silu(G)*U @ Wd^T (8 persistent 16x16 tiles/wave covers D).
__global__ __launch_bounds__(256) void moe_expert(
    const float* __restrict__ x,
    const float* __restrict__ wg,
    const float* __restrict__ wu,
    const float* __restrict__ wd,
    const int* __restrict__ counts,
    const int* __restrict__ lists,
    float* __restrict__ y,
    int N, int D, int H) {
  const int e = blockIdx.y;
  const int cnt = counts[e];
  const int t0 = blockIdx.x * MT;
  if (t0 >= cnt) return;
  const int nvalid = min(MT, cnt - t0);

  __shared__ int   s_entry[MT];
  __shared__ float sX[MT * SXD];   // full X tile, padded stride
  __shared__ float sH[MT * SHS];

  if ((int)threadIdx.x < MT) {
    int i = ((int)threadIdx.x < nvalid) ? (int)threadIdx.x : 0;
    s_entry[threadIdx.x] = lists[(size_t)e * N + t0 + i];
  }
  __syncthreads();

  // ---- Stage 0: async global->LDS DMA of the gathered X tile ----
  {
    const uint32_t base = lds_off(sX);
    const int rowq = D >> 2;  // float4s per row
    for (int i = threadIdx.x; i < MT * rowq; i += blockDim.x) {
      int r = i / rowq, c4 = i % rowq;
      int tok = s_entry[r] >> 1;
      async_copy_b128(base + (uint32_t)(r * SXD + c4 * 4) * 4u,
                      x + (size_t)tok * D + c4 * 4);
    }
    wait_async0();
    __syncthreads();
  }

  const int lane  = threadIdx.x & 31;
  const int wave  = threadIdx.x >> 5;
  const int mrow  = lane & 15;          // A/B row index per ISA 16x4 layout
  const int khalf = (lane >> 4) << 1;   // K = khalf, khalf+1
  const int cm    = (lane >> 4) << 3;   // C/D row base (0 or 8)
  const int cn    = lane & 15;          // C/D column

  const float* Wg = wg + (size_t)e * H * D;
  const float* Wu = wu + (size_t)e * H * D;
  const float* Wd = wd + (size_t)e * D * H;

  const int NJ = min(8, D >> 7);        // D/128 output tiles per wave (=8)
  v8f acc[8];
  for (int j = 0; j < 8; ++j) acc[j] = (v8f)0.f;

  const float* ax = &sX[mrow * SXD + khalf];

  for (int hc = 0; hc < H; hc += HC) {
    const int hw = hc + wave * 16;
    const float* bg = Wg + (size_t)(hw + mrow) * D + khalf;
    const float* bu = Wu + (size_t)(hw + mrow) * D + khalf;

    // ---- Stage 1: gate & up projections over full K, shared A operand ----
    v8f gac = (v8f)0.f, uac = (v8f)0.f;
    for (int k = 0; k < D; k += 4) {
      v2f a;  a.x  = ax[k]; a.y  = ax[k + 1];
      v2f b0; b0.x = bg[k]; b0.y = bg[k + 1];
      v2f b1; b1.x = bu[k]; b1.y = bu[k + 1];
      gac = wmma4(a, b0, gac);
      uac = wmma4(a, b1, uac);
    }

    // prefetch next H-chunk's weight rows while we do stage 2
    if (hc + HC < H) {
      __builtin_prefetch(bg + (size_t)HC * D, 0, 1);
      __builtin_prefetch(bu + (size_t)HC * D, 0, 1);
    }

    // h = silu(g) * u -> LDS (16 x HC)
    __syncthreads();  // previous stage-2 readers of sH are done
    for (int r = 0; r < 8; ++r) {
      float g = gac[r], u = uac[r];
      float hval = g * (1.f / (1.f + __expf(-g))) * u;
      sH[(cm + r) * SHS + wave * 16 + cn] = hval;
    }
    __syncthreads();

    // ---- Stage 2: acc[j] += h @ Wd^T over this H chunk ----
    const float* ah = &sH[mrow * SHS + khalf];
    for (int k = 0; k < HC; k += 4) {
      v2f a; a.x = ah[k]; a.y = ah[k + 1];
      for (int j = 0; j < NJ; ++j) {
        int db = (wave + (j << 3)) << 4;
        const float* bd = Wd + (size_t)(db + mrow) * H + hc + khalf + k;
        v2f b; b.x = bd[0]; b.y = bd[1];
        acc[j] = wmma4(a, b, acc[j]);
      }
    }
  }

  // scatter rows to per-(token,slot) slab; every (token,slot) written once
  for (int j = 0; j < NJ; ++j) {
    int db = (wave + (j << 3)) << 4;
    for (int r = 0; r < 8; ++r) {
      int m = cm + r;
      if (m < nvalid) {
        int entry = s_entry[m];
        int tok = entry >> 1, slot = entry & 1;
        y[((size_t)(tok << 1) + slot) * D + db + cn] = acc[j][r];
      }
    }
  }
}

__global__ void moe_combine(const float* __restrict__ y,
                            const float* __restrict__ wts,
                            float* __restrict__ out, size_t total, int D) {
  size_t i = (size_t)blockIdx.x * blockDim.x + threadIdx.x;
  if (i >= total) return;
  size_t n = i / D;
  size_t d = i - n * D;
  out[i] = wts[2 * n]     * y[(2 * n) * D + d] +
           wts[2 * n + 1] * y[(2 * n + 1) * D + d];
}

extern "C" void kernel_launch(void* const* d_in, const int* in_sizes, int n_in,
                              void* d_out, int out_size, void* d_ws, size_t ws_size,
                              hipStream_t stream) {
  const float* x  = (const float*)d_in[0];   // [B,T,D]
  const float* rw = (const float*)d_in[1];   // [E,D]
  const float* rb = (const float*)d_in[2];   // [E]
  const float* wg = (const float*)d_in[3];   // [E,H,D]
  const float* wu = (const float*)d_in[4];   // [E,H,D]
  const float* wd = (const float*)d_in[5];   // [E,D,H]
  float* out = (float*)d_out;

  const int E = in_sizes[2];
  const int D = in_sizes[1] / E;
  const int N = in_sizes[0] / D;
  const int H = in_sizes[3] / in_sizes[1];

  int*   counts = (int*)d_ws;
  int*   lists  = counts + 16;
  float* wts    = (float*)(lists + (size_t)E * N);
  float* y      = wts + (size_t)2 * N;
  // workspace use: 64B + E*N*4 + 2N*4 + 2N*D*4  (~67.4 MB for these shapes)

  moe_zero_counts<<<1, 64, 0, stream>>>(counts, E);
  moe_router<<<(N + 7) / 8, 256, 0, stream>>>(x, rw, rb, counts, lists, wts, N, D, E);
  dim3 grid((N + MT - 1) / MT, (unsigned)E);
  moe_expert<<<grid, 256, 0, stream>>>(x, wg, wu, wd, counts, lists, y, N, D, H);
  const size_t total = (size_t)N * D;
  moe_combine<<<(total + 255) / 256, 256, 0, stream>>>(y, wts, out, total, D);
}